// GlobalStyleEncoder_1305670058654
// MI455X (gfx1250) — compile-verified
//
#include <hip/hip_runtime.h>

typedef __attribute__((ext_vector_type(2))) float v2f;
typedef __attribute__((ext_vector_type(8))) float v8f;

#define NSEG 6
#define NCH  512
#define NPOS 16384   // 128*128
#define BINS 16

// ---------------------------------------------------------------------------
// CDNA5 async memory->LDS copy (ASYNCcnt-tracked), per-lane 8 bytes.
// LDS address = low 32 bits of the generic pointer (LDS aperture lives in
// addr[63:32]).
// ---------------------------------------------------------------------------
__device__ __forceinline__ void async_load_b64(const void* gsrc, void* lds) {
  asm volatile("global_load_async_to_lds_b64 %0, %1, off"
               :
               : "v"((unsigned)(unsigned long long)lds),
                 "v"((unsigned long long)gsrc)
               : "memory");
}

// ---------------------------------------------------------------------------
// Kernel 1: per-segment mask + rank (cumsum) + total count L.
// rankm[p] = rank among masked positions (row-major), or -1 if unmasked.
// ---------------------------------------------------------------------------
__global__ void seg_rank_kernel(const int* __restrict__ parsing,
                                int* __restrict__ rankm,
                                int* __restrict__ Ls) {
  __shared__ int tsum[256];
  const int s = blockIdx.x;
  const int t = threadIdx.x;
  const int per = NPOS / 256;           // 64 positions per thread
  const int base = t * per;

  int cnt = 0;
  for (int i = 0; i < per; ++i) {
    int p = base + i;
    int ph = p >> 7, pw = p & 127;      // 128x128 grid
    // nearest resize 256->128: src index (2*ph, 2*pw)
    int v = parsing[s * 65536 + (ph << 9) + (pw << 1)];
    cnt += (v != 0);
  }
  tsum[t] = cnt;
  __syncthreads();
  // Hillis-Steele inclusive scan over 256 thread counts
  for (int off = 1; off < 256; off <<= 1) {
    int add = (t >= off) ? tsum[t - off] : 0;
    __syncthreads();
    tsum[t] += add;
    __syncthreads();
  }
  int total = tsum[255];
  int r = tsum[t] - cnt;                // exclusive prefix
  for (int i = 0; i < per; ++i) {
    int p = base + i;
    int ph = p >> 7, pw = p & 127;
    int v = parsing[s * 65536 + (ph << 9) + (pw << 1)];
    int m = (v != 0);
    rankm[s * NPOS + p] = m ? r : -1;
    r += m;
  }
  if (t == 0) Ls[s] = total;
}

// ---------------------------------------------------------------------------
// Kernel 2: masked adaptive pooling as a [512 x 16384] x [16384 x 16] f32 GEMM
// using V_WMMA_F32_16X16X4_F32, with double-buffered async (memory->LDS)
// staging of the 16x64 x-tiles.  One wave per 16-channel M-tile; the 16-bin
// B-matrix is synthesized on the fly from rank/L (each lane owns bin n).
//
// A (16x4 f32) lane layout per ISA: lanes 0-15 -> M=0..15 with K={0,1},
// lanes 16-31 -> M=0..15 with K={2,3}.  D (16x16 f32): VGPR v, lane l holds
// D[v + 8*(l>=16)][l&15].
// ---------------------------------------------------------------------------
__global__ void pool_wmma_kernel(const float* __restrict__ xs,
                                 const int* __restrict__ rankm,
                                 const int* __restrict__ Ls,
                                 float* __restrict__ ys) {
  __shared__ __align__(16) float lx[2][4][16][66];  // double-buffered, padded
  __shared__ int lr[4][64];

  const int s    = blockIdx.y;
  const int wid  = threadIdx.x >> 5;
  const int lane = threadIdx.x & 31;
  const int mtile = blockIdx.x * 4 + wid;           // 0..31

  const float* xf = xs + ((size_t)s * NCH + (size_t)mtile * 16) * NPOS;
  const int*   rk = rankm + (size_t)s * NPOS;
  const int L = Ls[s];

  const int n    = lane & 15;                       // bin owned by this lane
  const int half = lane >> 4;
  int start = (n * L) >> 4;                         // floor(n*L/16)
  int end   = ((n + 1) * L + BINS - 1) >> 4;        // ceil((n+1)*L/16)
  int d = end - start; if (d < 1) d = 1;
  const float inv = 1.0f / (float)d;

  v8f acc = {0.f, 0.f, 0.f, 0.f, 0.f, 0.f, 0.f, 0.f};

  // prologue: async-stage chunk 0 into buffer 0 (16 rows x 64 floats)
#pragma unroll
  for (int rrow = 0; rrow < 16; ++rrow) {
    async_load_b64(xf + (size_t)rrow * NPOS + lane * 2,
                   &lx[0][wid][rrow][lane * 2]);
  }

  for (int kb = 0; kb < NPOS; kb += 64) {
    const int cur = (kb >> 6) & 1;
    const int nxt = cur ^ 1;
    const bool havePrefetch = (kb + 64) < NPOS;

    if (havePrefetch) {
      // WAR: prior iteration's ds reads of buffer `nxt` must be complete
      asm volatile("s_wait_dscnt 0x0" ::: "memory");
#pragma unroll
      for (int rrow = 0; rrow < 16; ++rrow) {
        async_load_b64(xf + (size_t)rrow * NPOS + (kb + 64) + lane * 2,
                       &lx[nxt][wid][rrow][lane * 2]);
      }
    }

    // ranks for the current chunk (wave-private LDS)
    lr[wid][lane]      = rk[kb + lane];
    lr[wid][lane + 32] = rk[kb + lane + 32];

    // wait for current chunk's async data: 16 prefetch loads (in order after
    // the current chunk's 16) may remain outstanding.
    if (havePrefetch) asm volatile("s_wait_asynccnt 0x10" ::: "memory");
    else              asm volatile("s_wait_asynccnt 0x0"  ::: "memory");

#pragma unroll
    for (int kk = 0; kk < 64; kk += 4) {
      // A: lane (m = lane&15, K pair selected by half)
      v2f a = *(const v2f*)&lx[cur][wid][lane & 15][kk + 2 * half];
      // B: lane holds B[2*half + {0,1}][n]
      int p0 = kk + 2 * half;
      int r0 = lr[wid][p0];
      int r1 = lr[wid][p0 + 1];
      v2f b;
      b.x = (r0 >= start && r0 < end) ? inv : 0.0f;   // r = -1 if unmasked
      b.y = (r1 >= start && r1 < end) ? inv : 0.0f;
      acc = __builtin_amdgcn_wmma_f32_16x16x4_f32(
          /*neg_a=*/false, a, /*neg_b=*/false, b,
          /*c_mod=*/(short)0, acc, /*reuse_a=*/false, /*reuse_b=*/false);
    }
  }

  // D layout: VGPR v, lane l -> row (v + 8*half), col n.  ys packed [s][c][bin]
#pragma unroll
  for (int v = 0; v < 8; ++v) {
    int row = mtile * 16 + v + 8 * half;
    ys[((size_t)s * NCH + row) * BINS + n] = acc[v];
  }
}

// ---------------------------------------------------------------------------
// Kernel 3: generic per-segment GEMV  y[s, yOff+row] = W[(s?)row,:] . x[s,:] + b
// One block (128 threads) per (segment,row); float4 coalesced loads.
// ---------------------------------------------------------------------------
__global__ void gemv_kernel(const float* __restrict__ W,
                            const float* __restrict__ bias,
                            const float* __restrict__ x,
                            float* __restrict__ y,
                            int M, int K, int wPerSeg, int bPerSeg,
                            int xStride, int xOff, int yStride, int yOff) {
  __shared__ float red[128];
  const int row = blockIdx.x % M;
  const int s   = blockIdx.x / M;
  const float* w  = W + ((size_t)(wPerSeg ? s : 0) * M + row) * K;
  const float* xv = x + (size_t)s * xStride + xOff;

  float sum = 0.0f;
  const int K4 = K >> 2;
  for (int j = threadIdx.x; j < K4; j += blockDim.x) {
    float4 a  = ((const float4*)w)[j];
    float4 bx = ((const float4*)xv)[j];
    sum += a.x * bx.x + a.y * bx.y + a.z * bx.z + a.w * bx.w;
  }
  red[threadIdx.x] = sum;
  __syncthreads();
  for (int off = 64; off > 0; off >>= 1) {
    if ((int)threadIdx.x < off) red[threadIdx.x] += red[threadIdx.x + off];
    __syncthreads();
  }
  if (threadIdx.x == 0) {
    float b = bPerSeg ? bias[(size_t)s * M + row] : bias[row];
    y[(size_t)s * yStride + yOff + row] = red[0] + b;
  }
}

// ---------------------------------------------------------------------------
extern "C" void kernel_launch(void* const* d_in, const int* in_sizes, int n_in,
                              void* d_out, int out_size, void* d_ws, size_t ws_size,
                              hipStream_t stream) {
  (void)in_sizes; (void)n_in; (void)out_size; (void)ws_size;

  const float* xs      = (const float*)d_in[0];
  const int*   parsing = (const int*)  d_in[1];
  const float* fcA_w   = (const float*)d_in[2];
  const float* fcA_b   = (const float*)d_in[3];
  const float* fcB_w   = (const float*)d_in[4];
  const float* fcB_b   = (const float*)d_in[5];
  const float* fcC_w   = (const float*)d_in[6];
  const float* fcC_b   = (const float*)d_in[7];
  const float* p1_w    = (const float*)d_in[8];
  const float* p1_b    = (const float*)d_in[9];
  const float* p2_w    = (const float*)d_in[10];
  const float* p2_b    = (const float*)d_in[11];
  const float* p3_w    = (const float*)d_in[12];
  const float* p3_b    = (const float*)d_in[13];
  float* out = (float*)d_out;

  // workspace layout (bytes)
  char*  ws    = (char*)d_ws;
  int*   rankm = (int*)  (ws + 0);        // 6*16384 ints  = 393216 B
  int*   Ls    = (int*)  (ws + 393216);   // 6 ints (padded to 256 B)
  float* ys    = (float*)(ws + 393472);   // 6*8192 f32    = 196608 B
  float* a1    = (float*)(ws + 590080);   // 6*2048 f32    =  49152 B
  float* a2    = (float*)(ws + 639232);   // 6*512  f32    =  12288 B

  seg_rank_kernel<<<NSEG, 256, 0, stream>>>(parsing, rankm, Ls);
  pool_wmma_kernel<<<dim3(8, NSEG), 128, 0, stream>>>(xs, rankm, Ls, ys);

  // fcA: [2048 x 8192] per-seg
  gemv_kernel<<<NSEG * 2048, 128, 0, stream>>>(fcA_w, fcA_b, ys, a1,
                                               2048, 8192, 1, 1, 8192, 0, 2048, 0);
  // fcB: [512 x 2048] per-seg
  gemv_kernel<<<NSEG * 512, 128, 0, stream>>>(fcB_w, fcB_b, a1, a2,
                                              512, 2048, 1, 1, 2048, 0, 512, 0);
  // fcC: [512 x 512] per-seg -> w0 directly into out[:, 0:512]
  gemv_kernel<<<NSEG * 512, 128, 0, stream>>>(fcC_w, fcC_b, a2, out,
                                              512, 512, 1, 1, 512, 0, 960, 0);
  // p1: shared [256 x 512], x = out[:,0:512], y = out[:,512:768]
  gemv_kernel<<<NSEG * 256, 128, 0, stream>>>(p1_w, p1_b, out, out,
                                              256, 512, 0, 0, 960, 0, 960, 512);
  // p2: shared [128 x 256], x = out[:,512:768], y = out[:,768:896]
  gemv_kernel<<<NSEG * 128, 128, 0, stream>>>(p2_w, p2_b, out, out,
                                              128, 256, 0, 0, 960, 512, 960, 768);
  // p3: shared [64 x 128], x = out[:,768:896], y = out[:,896:960]
  gemv_kernel<<<NSEG * 64, 128, 0, stream>>>(p3_w, p3_b, out, out,
                                             64, 128, 0, 0, 960, 768, 960, 896);
}